// pLaplacianConv_32736240730465
// MI455X (gfx1250) — compile-verified
//
#include <hip/hip_runtime.h>
#include <hip/hip_bf16.h>

#define D_FEAT 32
#define K_ITER 5
#define MU_F 0.01f

typedef __attribute__((ext_vector_type(2))) float v2f;
typedef __attribute__((ext_vector_type(8))) float v8f;

// ---------------------------------------------------------------------------
// Kernel 1: copy h_cur -> h_next (so scatter can atomically accumulate into
// it), and zero the 64 max-reduction slots.
// ---------------------------------------------------------------------------
__global__ __launch_bounds__(256) void plap_copy_init(const float* __restrict__ src,
                                                      float* __restrict__ dst,
                                                      unsigned* __restrict__ nmax_slots,
                                                      int n4) {
    int i = blockIdx.x * blockDim.x + threadIdx.x;
    if (i < n4) ((float4*)dst)[i] = ((const float4*)src)[i];
    if (blockIdx.x == 0 && threadIdx.x < 64) nmax_slots[threadIdx.x] = 0u;
}

// ---------------------------------------------------------------------------
// Kernel 2: per-edge norms via V_WMMA_F32_16X16X4_F32.
// Each wave handles 16 edges: stage squared diffs (16x32 f32) in LDS, then
// 8 WMMAs with B = ones(4x16) give C[i][j] = sum_k sq[i][k] (exact f32 FMA).
// Lanes 0/16 hold the N=0 column -> the 16 norms. Block max -> 64 slots.
// ---------------------------------------------------------------------------
__global__ __launch_bounds__(256) void plap_norms(const float* __restrict__ h,
                                                  const long long* __restrict__ eidx,
                                                  float* __restrict__ norms,
                                                  unsigned* __restrict__ nmax_slots,
                                                  int E) {
    __shared__ float sq[8][16 * D_FEAT];   // 16 KB: 8 waves x 16 edges x 32 feats
    __shared__ float wmax[16];

    const int lane = threadIdx.x & 31;
    const int wave = threadIdx.x >> 5;
    const long long ebase = (long long)blockIdx.x * 128 + (long long)wave * 16;

    // Stage squared diffs: 4 edges per sub-step, 8 lanes/edge, float4 each.
    #pragma unroll
    for (int s = 0; s < 4; ++s) {
        const int el = s * 4 + (lane >> 3);        // 0..15 edge-in-wave
        const long long e = ebase + el;
        const int f = (lane & 7) * 4;
        float4 d4 = make_float4(0.f, 0.f, 0.f, 0.f);
        if (e < E) {
            const int r = (int)eidx[e];
            const int c = (int)eidx[(long long)E + e];
            const float4 a = *(const float4*)(h + (long long)r * D_FEAT + f);
            const float4 b = *(const float4*)(h + (long long)c * D_FEAT + f);
            d4.x = a.x - b.x; d4.y = a.y - b.y; d4.z = a.z - b.z; d4.w = a.w - b.w;
        }
        float* p = &sq[wave][el * D_FEAT + f];
        p[0] = d4.x * d4.x; p[1] = d4.y * d4.y; p[2] = d4.z * d4.z; p[3] = d4.w * d4.w;
    }
    __syncthreads();

    // A-matrix 16x4 f32 layout: lanes 0-15 -> (M=lane, K=0,1); lanes 16-31 ->
    // (M=lane-16, K=2,3). Accumulate 8 chunks of K=4 over the 32 features.
    const int M = lane & 15;
    const int koff = (lane >> 4) * 2;
    v8f c = {};
    v2f bones = {1.0f, 1.0f};
    #pragma unroll
    for (int t = 0; t < 8; ++t) {
        const float* p = &sq[wave][M * D_FEAT + t * 4 + koff];
        v2f a; a[0] = p[0]; a[1] = p[1];
        c = __builtin_amdgcn_wmma_f32_16x16x4_f32(false, a, false, bones,
                                                  (short)0, c, false, false);
    }

    // C layout: VGPR j, lane 0 = C[M=j][N=0]; lane 16 = C[M=j+8][N=0].
    float lmax = 0.f;
    if (lane == 0 || lane == 16) {
        const int eoff = (lane >> 4) * 8;
        #pragma unroll
        for (int j = 0; j < 8; ++j) {
            const long long e = ebase + eoff + j;
            if (e < E) {
                float nv = sqrtf(c[j]);
                nv = fmaxf(nv, 1e-6f);
                norms[e] = nv;
                lmax = fmaxf(lmax, nv);
            }
        }
        wmax[wave * 2 + (lane >> 4)] = lmax;
    }
    __syncthreads();
    if (threadIdx.x == 0) {
        float m = 0.f;
        #pragma unroll
        for (int j = 0; j < 16; ++j) m = fmaxf(m, wmax[j]);
        // norms are >= 1e-6 > 0, so uint ordering == float ordering
        atomicMax(&nmax_slots[blockIdx.x & 63], __float_as_uint(m));
    }
}

// ---------------------------------------------------------------------------
// Kernel 3: reduce the 64 slots to the final max scalar.
// ---------------------------------------------------------------------------
__global__ void plap_reduce_max(const unsigned* __restrict__ slots,
                                float* __restrict__ out) {
    __shared__ unsigned sm[64];
    sm[threadIdx.x] = slots[threadIdx.x];
    __syncthreads();
    if (threadIdx.x == 0) {
        unsigned m = 0u;
        #pragma unroll
        for (int i = 0; i < 64; ++i) m = (sm[i] > m) ? sm[i] : m;
        out[0] = __uint_as_float(m);
    }
}

// ---------------------------------------------------------------------------
// Kernel 4: scatter. h_next[row] += MU * (norm/nmax)^(-0.5) * (h[row]-h[col])
// 8 lanes per edge, 4 hardware f32 atomics per lane (L2-resident target).
// ---------------------------------------------------------------------------
__global__ __launch_bounds__(256) void plap_scatter(const float* __restrict__ h,
                                                    const long long* __restrict__ eidx,
                                                    const float* __restrict__ norms,
                                                    const float* __restrict__ nmax,
                                                    float* __restrict__ hnext,
                                                    int E) {
    const long long tid = (long long)blockIdx.x * blockDim.x + threadIdx.x;
    const int e = (int)(tid >> 3);
    if (e >= E) return;
    const int f = ((int)tid & 7) * 4;
    const int r = (int)eidx[e];
    const int c = (int)eidx[(long long)E + e];
    const float4 a = *(const float4*)(h + (long long)r * D_FEAT + f);
    const float4 b = *(const float4*)(h + (long long)c * D_FEAT + f);
    // w = (norm/nmax)^(P-2), P-2 = -0.5  ->  w = sqrt(nmax/norm)
    const float s = MU_F * sqrtf(nmax[0] / norms[e]);
    float* dst = hnext + (long long)r * D_FEAT + f;
    unsafeAtomicAdd(dst + 0, s * (a.x - b.x));
    unsafeAtomicAdd(dst + 1, s * (a.y - b.y));
    unsafeAtomicAdd(dst + 2, s * (a.z - b.z));
    unsafeAtomicAdd(dst + 3, s * (a.w - b.w));
}

extern "C" void kernel_launch(void* const* d_in, const int* in_sizes, int n_in,
                              void* d_out, int out_size, void* d_ws, size_t ws_size,
                              hipStream_t stream) {
    const float* h_in = (const float*)d_in[0];
    const long long* eidx = (const long long*)d_in[1];
    const int N = in_sizes[0] / D_FEAT;
    const int E = in_sizes[1] / 2;
    const int NF = N * D_FEAT;

    // Workspace layout: two ping-pong h buffers, per-edge norms, 64 max slots,
    // final max scalar. Total ~32 MB for N=100k, E=1.6M.
    float* W0 = (float*)d_ws;
    float* W1 = W0 + NF;
    float* norms = W1 + NF;
    unsigned* slots = (unsigned*)(norms + E);
    float* nmaxf = (float*)(slots + 64);

    const int copyBlocks = (NF / 4 + 255) / 256;
    const int normBlocks = (E + 127) / 128;
    const long long sthreads = (long long)E * 8;
    const int scatBlocks = (int)((sthreads + 255) / 256);

    const float* cur = h_in;
    for (int it = 0; it < K_ITER; ++it) {
        float* next = (it == K_ITER - 1) ? (float*)d_out : ((it & 1) ? W1 : W0);
        plap_copy_init<<<copyBlocks, 256, 0, stream>>>(cur, next, slots, NF / 4);
        plap_norms<<<normBlocks, 256, 0, stream>>>(cur, eidx, norms, slots, E);
        plap_reduce_max<<<1, 64, 0, stream>>>(slots, nmaxf);
        plap_scatter<<<scatBlocks, 256, 0, stream>>>(cur, eidx, norms, nmaxf, next, E);
        cur = next;
    }
}